// MTPWorker_87471303950911
// MI455X (gfx1250) — compile-verified
//
#include <hip/hip_runtime.h>
#include <hip/hip_bf16.h>
#include <stdint.h>

// Problem constants (match reference)
#define KD     3
#define BD     64
#define VD     128000
#define HD     7168
#define MAXRD  128

#define THREADS 256
#define SEGS    5                               // segments per vocab row
#define NBUF    8                               // async pipeline depth (LDS buffers)
#define CHUNKS  ((VD / 4) / THREADS / SEGS)     // 25 f4-chunks per segment (exact)

// ---- CDNA5 async global->LDS primitives (CDNA5 ISA) ----

template <int N>
__device__ __forceinline__ void wait_asynccnt_le() {
  asm volatile("s_wait_asynccnt %0" :: "n"(N) : "memory");
}

__device__ __forceinline__ void wait_dscnt0() {
  asm volatile("s_wait_dscnt 0x0" ::: "memory");
}

// GVS-mode async load: copies from global saddr + vaddr into LDS at vdst. 16B per lane, ASYNCcnt++.
__device__ __forceinline__ void async_load_b128(unsigned lds_addr, unsigned byte_off,
                                                unsigned long long base) {
  asm volatile("global_load_async_to_lds_b128 %0, %1, %2"
               :: "v"(lds_addr), "v"(byte_off), "s"(base)
               : "memory");
}

// ---------------------------------------------------------------------------
// Kernel 1: streaming row-argmax over a vocab-row segment, optionally fusing
// the passthrough copy of the row to the output. One block per (row, segment).
// ---------------------------------------------------------------------------
template <bool COPY>
__global__ __launch_bounds__(THREADS)
void rowmax_stream_kernel(const float* __restrict__ in,
                          float* __restrict__ out_copy,
                          float* __restrict__ part_val,
                          int* __restrict__ part_idx) {
  __shared__ float4 sbuf[NBUF][THREADS];
  __shared__ float wval[THREADS / 32];
  __shared__ int   widx[THREADS / 32];

  const int tid = threadIdx.x;
  const int bx  = blockIdx.x;
  const int r   = bx / SEGS;
  const int s   = bx % SEGS;

  const unsigned long long base =
      (unsigned long long)(const void*)(in + (size_t)r * VD);
  float4* out_row = COPY ? (float4*)(out_copy + (size_t)r * VD) : nullptr;

  // Generic->LDS truncation: low 32 bits of a generic LDS pointer are the
  // LDS byte offset (ISA aperture rules). Buffers are contiguous.
  const unsigned lds0 = (unsigned)(uintptr_t)&sbuf[0][tid];
  const unsigned lds_stride = (unsigned)(THREADS * sizeof(float4));

  const int c0 = s * CHUNKS;  // first f4-chunk of this segment within the row

  // Prologue: fill the pipeline NBUF deep.
#pragma unroll
  for (int p = 0; p < NBUF; ++p) {
    async_load_b128(lds0 + (unsigned)p * lds_stride,
                    (unsigned)(((c0 + p) * THREADS + tid) * 16), base);
  }

  float bv = -3.402823466e38f;
  int   bi = 0;

#pragma unroll 1
  for (int c = 0; c < CHUNKS; ++c) {
    // Ensure chunk c has landed in LDS (loads retire in order per wave).
    if (c <= CHUNKS - NBUF) {
      wait_asynccnt_le<NBUF - 1>();
    } else if (c == CHUNKS - NBUF + 1) {
      wait_asynccnt_le<0>();  // drain: all remaining chunks are resident
    }

    const float4 v = sbuf[c & (NBUF - 1)][tid];
    const int e = ((c0 + c) * THREADS + tid) * 4;
    if (v.x > bv) { bv = v.x; bi = e;     }
    if (v.y > bv) { bv = v.y; bi = e + 1; }
    if (v.z > bv) { bv = v.z; bi = e + 2; }
    if (v.w > bv) { bv = v.w; bi = e + 3; }
    if (COPY) out_row[(c0 + c) * THREADS + tid] = v;

    // Refill the buffer we just consumed.
    if (c < CHUNKS - NBUF) {
      wait_dscnt0();  // our ds read must complete before async overwrite
      async_load_b128(lds0 + (unsigned)(c & (NBUF - 1)) * lds_stride,
                      (unsigned)(((c0 + c + NBUF) * THREADS + tid) * 16), base);
    }
  }

  // Wave32 reduction (first-occurrence tie-break => lower index wins).
#pragma unroll
  for (int off = 16; off > 0; off >>= 1) {
    float ov = __shfl_down(bv, off, 32);
    int   oi = __shfl_down(bi, off, 32);
    if (ov > bv || (ov == bv && oi < bi)) { bv = ov; bi = oi; }
  }
  const int wid = tid >> 5, lane = tid & 31;
  if (lane == 0) { wval[wid] = bv; widx[wid] = bi; }
  __syncthreads();
  if (tid == 0) {
#pragma unroll
    for (int w = 1; w < THREADS / 32; ++w) {
      float ov = wval[w]; int oi = widx[w];
      if (ov > bv || (ov == bv && oi < bi)) { bv = ov; bi = oi; }
    }
    part_val[r * SEGS + s] = bv;
    part_idx[r * SEGS + s] = bi;
  }
}

// ---------------------------------------------------------------------------
// Kernel 2: combine per-segment partials into final argmax tokens.
// ---------------------------------------------------------------------------
__global__ void reduce_rows_kernel(const float* __restrict__ pv_l, const int* __restrict__ pi_l,
                                   const float* __restrict__ pv_d, const int* __restrict__ pi_d,
                                   int* __restrict__ tgt, int* __restrict__ dtok) {
  const int t = threadIdx.x;
  const int n_l = BD * (KD + 1);
  const int n_d = BD * KD;
  if (t < n_l) {
    float bv = pv_l[t * SEGS]; int bi = pi_l[t * SEGS];
    for (int s = 1; s < SEGS; ++s) {
      float ov = pv_l[t * SEGS + s]; int oi = pi_l[t * SEGS + s];
      if (ov > bv || (ov == bv && oi < bi)) { bv = ov; bi = oi; }
    }
    tgt[t] = bi;
  } else if (t < n_l + n_d) {
    const int q = t - n_l;
    float bv = pv_d[q * SEGS]; int bi = pi_d[q * SEGS];
    for (int s = 1; s < SEGS; ++s) {
      float ov = pv_d[q * SEGS + s]; int oi = pi_d[q * SEGS + s];
      if (ov > bv || (ov == bv && oi < bi)) { bv = ov; bi = oi; }
    }
    dtok[q] = bi;
  }
}

// ---------------------------------------------------------------------------
// Kernel 3: acceptance logic + token-pool update + small outputs.
// ---------------------------------------------------------------------------
__global__ __launch_bounds__(THREADS)
void accept_finalize_kernel(const int* __restrict__ tgt, const int* __restrict__ dtok,
                            const int* __restrict__ draft_tokens,
                            const int* __restrict__ slot_ids,
                            const int* __restrict__ past_tpool,
                            float* __restrict__ out_acc, float* __restrict__ out_num,
                            float* __restrict__ out_draft, float* __restrict__ out_next,
                            float* __restrict__ out_tpool,
                            int* __restrict__ ws_nacc, int* __restrict__ ws_smap) {
  const int tid = threadIdx.x;
  for (int r = tid; r < MAXRD; r += THREADS) ws_smap[r] = -1;
  for (int i = tid; i < MAXRD * KD; i += THREADS) out_tpool[i] = (float)past_tpool[i];
  __syncthreads();

  if (tid < BD) {
    const int b = tid;
    const int slot = slot_ids[b];
    ws_smap[slot] = b;

    int t[KD + 1];
#pragma unroll
    for (int j = 0; j <= KD; ++j) t[j] = tgt[b * (KD + 1) + j];

    int n = 1, ok = 1;
#pragma unroll
    for (int k = 0; k < KD; ++k) {
      ok &= (t[k] == draft_tokens[b * KD + k]) ? 1 : 0;
      n += ok;
    }
    ws_nacc[b] = n;

#pragma unroll
    for (int j = 0; j <= KD; ++j) out_acc[b * (KD + 1) + j] = (float)t[j];
    out_num[b] = (float)n;

#pragma unroll
    for (int k = 0; k < KD; ++k) {
      const int g = n + k;  // gidx in [1, 2K]
      const int v = (g < KD) ? past_tpool[slot * KD + g] : t[g - KD];
      out_tpool[slot * KD + k] = (float)v;
      const int d = dtok[b * KD + k];
      out_draft[b * KD + k] = (float)d;
      out_next[b * (KD + 1) + 1 + k] = (float)d;
    }
    out_next[b * (KD + 1)] = (float)t[n - 1];
  }
}

// ---------------------------------------------------------------------------
// Kernel 4: hidden pool gather/scatter. One block per (slot, k) row of H.
// ---------------------------------------------------------------------------
__global__ __launch_bounds__(THREADS)
void hidden_pool_kernel(const float* __restrict__ past_hpool,
                        const float* __restrict__ hidden_states,
                        const int* __restrict__ ws_nacc,
                        const int* __restrict__ ws_smap,
                        float* __restrict__ out_hpool) {
  const int bx = blockIdx.x;            // [0, MAXRD*KD)
  const int r = bx / KD, k = bx % KD;
  const int b = ws_smap[r];

  const float* src;
  if (b < 0) {
    src = past_hpool + (size_t)bx * HD;  // slot untouched: passthrough copy
  } else {
    const int g = ws_nacc[b] + k;        // index into concat(past_K, hidden_{K+1})
    src = (g < KD) ? past_hpool + ((size_t)r * KD + g) * HD
                   : hidden_states + ((size_t)b * (KD + 1) + (g - KD)) * HD;
  }
  float* dst = out_hpool + (size_t)bx * HD;

  const float4* s4 = (const float4*)src;
  float4* d4 = (float4*)dst;
  for (int i = threadIdx.x; i < HD / 4; i += THREADS) d4[i] = s4[i];  // 7 iters
}

// ---------------------------------------------------------------------------
extern "C" void kernel_launch(void* const* d_in, const int* in_sizes, int n_in,
                              void* d_out, int out_size, void* d_ws, size_t ws_size,
                              hipStream_t stream) {
  (void)in_sizes; (void)n_in; (void)out_size; (void)ws_size;

  const float* logits       = (const float*)d_in[0];
  const float* hidden       = (const float*)d_in[1];
  const int*   draft_tokens = (const int*)d_in[2];
  const int*   slot_ids     = (const int*)d_in[3];
  const float* past_hpool   = (const float*)d_in[4];
  const int*   past_tpool   = (const int*)d_in[5];
  const float* draft_logits = (const float*)d_in[6];
  float* out = (float*)d_out;

  // Output layout: concat of reference tuple, flat, row-major.
  const size_t O_LOG = 0;
  const size_t O_ACC = O_LOG + (size_t)BD * (KD + 1) * VD;
  const size_t O_NUM = O_ACC + (size_t)BD * (KD + 1);
  const size_t O_DRA = O_NUM + (size_t)BD;
  const size_t O_NXT = O_DRA + (size_t)BD * KD;
  const size_t O_HPO = O_NXT + (size_t)BD * (KD + 1);
  const size_t O_TPO = O_HPO + (size_t)MAXRD * KD * HD;

  // Workspace layout.
  char* w = (char*)d_ws;
  float* pv_l = (float*)w; w += sizeof(float) * BD * (KD + 1) * SEGS;
  int*   pi_l = (int*)w;   w += sizeof(int)   * BD * (KD + 1) * SEGS;
  float* pv_d = (float*)w; w += sizeof(float) * BD * KD * SEGS;
  int*   pi_d = (int*)w;   w += sizeof(int)   * BD * KD * SEGS;
  int*   tgt  = (int*)w;   w += sizeof(int)   * BD * (KD + 1);
  int*   dtok = (int*)w;   w += sizeof(int)   * BD * KD;
  int*   nacc = (int*)w;   w += sizeof(int)   * BD;
  int*   smap = (int*)w;   w += sizeof(int)   * MAXRD;

  // 1) argmax over target logits, fused with the passthrough copy (read once).
  rowmax_stream_kernel<true><<<BD * (KD + 1) * SEGS, THREADS, 0, stream>>>(
      logits, out + O_LOG, pv_l, pi_l);
  // 2) argmax over draft logits (no copy-out).
  rowmax_stream_kernel<false><<<BD * KD * SEGS, THREADS, 0, stream>>>(
      draft_logits, nullptr, pv_d, pi_d);
  // 3) combine segment partials -> tokens.
  reduce_rows_kernel<<<1, 512, 0, stream>>>(pv_l, pi_l, pv_d, pi_d, tgt, dtok);
  // 4) acceptance + token pool + small outputs.
  accept_finalize_kernel<<<1, THREADS, 0, stream>>>(
      tgt, dtok, draft_tokens, slot_ids, past_tpool,
      out + O_ACC, out + O_NUM, out + O_DRA, out + O_NXT, out + O_TPO,
      nacc, smap);
  // 5) hidden pool gather/scatter.
  hidden_pool_kernel<<<MAXRD * KD, THREADS, 0, stream>>>(
      past_hpool, hidden, nacc, smap, out + O_HPO);
}